// BackRazorSoftmaxMatMul_65755949302073
// MI455X (gfx1250) — compile-verified
//
#include <hip/hip_runtime.h>

// ---------------------------------------------------------------------------
// softmax(x1, dim=-1) @ x2^T   for x1:[2,16,2048,2048] f32, x2:[2,16,64,2048]
// One workgroup = one (b,h) pair + 16 query rows. 128 threads = 4 wave32.
// Phase 0: stage 16x2048 score tile into LDS via GLOBAL_LOAD_ASYNC_TO_LDS_B128
//          (per-lane LDS dest -> keeps the padded stride-2052 layout),
//          fallback to regular loads + ds_store if the builtin is absent.
// Phase 1: row max, in-place exp2 (v_exp_f32), row sum.
// Phase 2: each wave computes a 16x16 output tile with V_WMMA_F32_16X16X4_F32
//          accumulating over K=2048, then scales by 1/rowsum at store.
// ---------------------------------------------------------------------------

typedef float v2f __attribute__((ext_vector_type(2)));
typedef float v8f __attribute__((ext_vector_type(8)));
typedef int   v4i __attribute__((ext_vector_type(4)));

typedef __attribute__((address_space(1))) v4i* as1_v4i_p;  // global int4*
typedef __attribute__((address_space(3))) v4i* as3_v4i_p;  // LDS int4*

#if __has_builtin(__builtin_amdgcn_global_load_async_to_lds_b128)
#define HAVE_ASYNC_LDS 1
#endif

constexpr int S_DIM    = 2048;  // sequence length (both q and k)
constexpr int DH       = 64;    // head dim
constexpr int ROWS     = 16;    // query rows per workgroup (= WMMA M)
constexpr int STRIDE   = 2052;  // LDS row stride in floats (2048 + 4 -> 4-bank row skew)
constexpr int NTHREADS = 128;   // 4 wave32
constexpr float LOG2E  = 1.44269504088896340736f;

__global__ __launch_bounds__(NTHREADS)
void softmax_matmul_wmma(const float* __restrict__ x1,
                         const float* __restrict__ x2,
                         float* __restrict__ out) {
  extern __shared__ float smem[];
  float* probs   = smem;                   // ROWS * STRIDE floats (~128.25 KB)
  float* scratch = smem + ROWS * STRIDE;   // NTHREADS floats (partial reductions)
  float* stats   = scratch + NTHREADS;     // [0..15] row max, [16..31] 1/rowsum

  const int t  = threadIdx.x;
  const int bh = blockIdx.x >> 7;               // 128 row-tiles per (b,h)
  const int q0 = (blockIdx.x & 127) * ROWS;

  const float* x1t = x1 + ((size_t)bh * S_DIM + q0) * S_DIM;  // 16 contiguous rows

  // ---- stage x1 tile into LDS (contiguous 128 KB block, 16 B granularity) ----
#ifdef HAVE_ASYNC_LDS
  for (int i = t; i < ROWS * S_DIM / 4; i += NTHREADS) {
    const int row = i >> 9;      // 512 float4 per source row
    const int c4  = i & 511;
    const float* gsrc = x1t + (size_t)i * 4;
    float*       ldst = probs + row * STRIDE + c4 * 4;
    __builtin_amdgcn_global_load_async_to_lds_b128(
        (as1_v4i_p)gsrc, (as3_v4i_p)ldst, /*offset=*/0, /*cpol=*/0);
  }
#if __has_builtin(__builtin_amdgcn_s_wait_asynccnt)
  __builtin_amdgcn_s_wait_asynccnt(0);
#else
  asm volatile("s_wait_asynccnt 0" ::: "memory");
#endif
#else
  for (int i = t; i < ROWS * S_DIM / 4; i += NTHREADS) {
    const float4 v  = ((const float4*)x1t)[i];
    const int  row  = i >> 9;
    const int  c4   = i & 511;
    *(float4*)(probs + row * STRIDE + c4 * 4) = v;
  }
#endif
  __syncthreads();

  // ---- per-row max (8 threads per row, LDS reduce) ----
  const int r   = t & 15;
  const int sub = t >> 4;          // 0..7
  {
    float m = -3.402823466e38f;
    for (int c4 = sub; c4 < S_DIM / 4; c4 += 8) {
      const float4 v = *(const float4*)(probs + r * STRIDE + c4 * 4);
      m = fmaxf(m, fmaxf(fmaxf(v.x, v.y), fmaxf(v.z, v.w)));
    }
    scratch[r * 8 + sub] = m;
  }
  __syncthreads();
  if (t < ROWS) {
    float m = scratch[t * 8];
    #pragma unroll
    for (int j = 1; j < 8; ++j) m = fmaxf(m, scratch[t * 8 + j]);
    stats[t] = m;
  }
  __syncthreads();

  // ---- exp in place (v_exp_f32) + per-row sum ----
  {
    const float m = stats[r];
    float s = 0.0f;
    for (int c4 = sub; c4 < S_DIM / 4; c4 += 8) {
      float4 v = *(float4*)(probs + r * STRIDE + c4 * 4);
      v.x = __builtin_amdgcn_exp2f((v.x - m) * LOG2E);
      v.y = __builtin_amdgcn_exp2f((v.y - m) * LOG2E);
      v.z = __builtin_amdgcn_exp2f((v.z - m) * LOG2E);
      v.w = __builtin_amdgcn_exp2f((v.w - m) * LOG2E);
      *(float4*)(probs + r * STRIDE + c4 * 4) = v;
      s += (v.x + v.y) + (v.z + v.w);
    }
    scratch[r * 8 + sub] = s;
  }
  __syncthreads();
  if (t < ROWS) {
    float s = 0.0f;
    #pragma unroll
    for (int j = 0; j < 8; ++j) s += scratch[t * 8 + j];
    stats[16 + t] = 1.0f / s;
  }
  __syncthreads();

  // ---- WMMA phase: P(16 x 2048) @ x2[d, k]^T, wave w owns d = 16w..16w+15 ----
  const int wave = t >> 5;
  const int lane = t & 31;
  const int ln   = lane & 15;      // M for A-frag / N for B-frag
  const int hi   = lane >> 4;      // K half-select (ISA 16x4 f32 layout)
  const int d    = wave * 16 + ln;

  // A frag: lane holds (M=ln, K = k0 + 2*hi + {0,1})     [ISA 7.12.2]
  // B frag: lane holds (N=ln, K = k0 + 2*hi + {0,1})  -> x2 row d, k-contiguous
  const float* Arow = probs + ln * STRIDE + 2 * hi;
  const float* Brow = x2 + ((size_t)bh * DH + d) * S_DIM + 2 * hi;

  v8f acc = {0.f, 0.f, 0.f, 0.f, 0.f, 0.f, 0.f, 0.f};
  #pragma unroll 8
  for (int k0 = 0; k0 < S_DIM; k0 += 4) {
    v2f a = *(const v2f*)(Arow + k0);   // ds_load (bank-conflict-free)
    v2f b = *(const v2f*)(Brow + k0);   // global_load_b64, L2-resident
    acc = __builtin_amdgcn_wmma_f32_16x16x4_f32(
        /*neg_a=*/false, a, /*neg_b=*/false, b,
        /*c_mod=*/(short)0, acc, /*reuse_a=*/false, /*reuse_b=*/false);
  }

  // D layout: VGPR j <-> row M = j + 8*hi, col N = ln.  Scale by 1/rowsum[M].
  const float4 i0 = *(const float4*)(stats + 16 + hi * 8);
  const float4 i1 = *(const float4*)(stats + 16 + hi * 8 + 4);
  const float inv[8] = {i0.x, i0.y, i0.z, i0.w, i1.x, i1.y, i1.z, i1.w};

  float* orow = out + ((size_t)bh * S_DIM + q0 + hi * 8) * DH + d;
  #pragma unroll
  for (int j = 0; j < 8; ++j) {
    orow[(size_t)j * DH] = acc[j] * inv[j];
  }
}

extern "C" void kernel_launch(void* const* d_in, const int* in_sizes, int n_in,
                              void* d_out, int out_size, void* d_ws, size_t ws_size,
                              hipStream_t stream) {
  const float* x1 = (const float*)d_in[0];   // [2,16,2048,2048] f32
  const float* x2 = (const float*)d_in[1];   // [2,16,64,2048]  f32
  float* out      = (float*)d_out;           // [2,16,2048,64]  f32

  const int num_bh    = 2 * 16;
  const int row_tiles = S_DIM / ROWS;        // 128
  dim3 grid(num_bh * row_tiles);             // 4096 workgroups
  dim3 block(NTHREADS);
  const size_t shmem = (size_t)(ROWS * STRIDE + NTHREADS + 32) * sizeof(float);

  softmax_matmul_wmma<<<grid, block, shmem, stream>>>(x1, x2, out);
}